// WeightedAtomLayer_6047313953396
// MI455X (gfx1250) — compile-verified
//
#include <hip/hip_runtime.h>

// Reference: y[k,n,o] = sum_d x[k,n,d]*w[k,o,d]; out = tanh(y.reshape(N,256)[perm])
// K=16 heads, N_PER=16384 rows/head, D_IN=D_OUT=256, all f32.
//
// MI455X (gfx1250, wave32) strategy:
//  - kernel 1: inverse permutation into d_ws
//  - kernel 2: per (head, 128-row block) WG GEMM with V_WMMA_F32_16X16X4_F32
//      * W[k] staged in LDS as K-PAIRED float2 elements: Wp[s2][o] = {W[o][2s2], W[o][2s2+1]}
//        -> each B fragment is ONE ds_load_b64, no register shuffles
//      * padded row stride 273 float2 (546 dwords, 546%64=34): conflict-free fill,
//        ~conflict-free fragment loads
//      * A (16 rows x 256) in 128 VGPRs per wave, native A-fragment layout
//      * fused tanh + permuted row scatter epilogue

typedef __attribute__((ext_vector_type(2))) float v2f;
typedef __attribute__((ext_vector_type(8))) float v8f;

#define KHEADS 16
#define NPER   16384
#define DIM    256
#define S2DIM  128          // number of K-pairs (DIM/2)
#define LDSW2  273          // padded row stride in float2 units

__global__ void invperm_kernel(const int* __restrict__ perm,
                               int* __restrict__ inv, int n) {
  int i = blockIdx.x * blockDim.x + threadIdx.x;
  if (i < n) inv[perm[i]] = i;
}

__global__ __launch_bounds__(256)
void gemm_tanh_perm_kernel(const float* __restrict__ x,   // [K, NPER, DIM]
                           const float* __restrict__ w,   // [K, DIM, DIM] (o major, d contiguous)
                           const int*   __restrict__ inv, // [K*NPER] inverse permutation
                           float*       __restrict__ out) // [K*NPER, DIM]
{
  extern __shared__ v2f Wp[];            // [S2DIM][LDSW2]: Wp[s2*LDSW2+o] = {W[o][2s2],W[o][2s2+1]}
  const int tid  = threadIdx.x;
  const int lane = tid & 31;
  const int wave = tid >> 5;
  const int k    = blockIdx.y;
  const int rowbase = blockIdx.x * 128 + wave * 16;   // 8 waves x 16 rows

  const int half = lane >> 4;            // 0 -> K%4 in {0,1}, 1 -> {2,3}
  const int m    = lane & 15;            // A row in tile / B,C column in tile

  // ---- A operand: native 16x4 f32 A-fragment layout, 64 v2f per lane ----
  // lane (half,m) holds row (rowbase+m), elements d = 4*s + 2*half + {0,1}
  const float* xrow = x + (size_t)(k * NPER + rowbase + m) * DIM + 2 * half;
  v2f a[64];
#pragma unroll
  for (int s = 0; s < 64; ++s)
    a[s] = *(const v2f*)(xrow + 4 * s);

  // ---- cooperative W[k] -> LDS, K-paired transpose ----
  // global: consecutive lanes read consecutive float2 (coalesced b64)
  // LDS: lanes walk s2, dword-bank stride 34 -> all 64 banks, conflict-free
  const v2f* wk2 = (const v2f*)(w + (size_t)k * DIM * DIM);
#pragma unroll 4
  for (int i = tid; i < S2DIM * DIM; i += 256) {
    int s2 = i & (S2DIM - 1);
    int o  = i >> 7;
    Wp[s2 * LDSW2 + o] = wk2[o * S2DIM + s2];
  }
  __syncthreads();

  // ---- permuted destination rows (C layout: VGPR j = row j + 8*half) ----
  int orow[8];
#pragma unroll
  for (int j = 0; j < 8; ++j)
    orow[j] = inv[k * NPER + rowbase + j + 8 * half];

  // ---- main loop: 16 N-tiles in pairs (2 independent accumulator chains) ----
#pragma unroll 1
  for (int ntp = 0; ntp < 8; ++ntp) {
    const int nt0 = 2 * ntp;
    v8f acc0 = {};
    v8f acc1 = {};
    // B fragment for step s: Wp[(2s+half)*LDSW2 + nt*16 + m] -> single b64 load
    const v2f* b0 = Wp + (size_t)half * LDSW2 + nt0 * 16 + m;
    const v2f* b1 = b0 + 16;
#pragma unroll
    for (int s = 0; s < 64; ++s) {
      v2f bb0 = b0[2 * s * LDSW2];
      v2f bb1 = b1[2 * s * LDSW2];
      acc0 = __builtin_amdgcn_wmma_f32_16x16x4_f32(false, a[s], false, bb0,
                                                   (short)0, acc0, false, false);
      acc1 = __builtin_amdgcn_wmma_f32_16x16x4_f32(false, a[s], false, bb1,
                                                   (short)0, acc1, false, false);
    }
    // ---- epilogue: tanh + permuted row scatter ----
#pragma unroll
    for (int j = 0; j < 8; ++j) {
      float* p = out + (size_t)orow[j] * DIM + m;
      p[(nt0 + 0) * 16] = tanhf(acc0[j]);
      p[(nt0 + 1) * 16] = tanhf(acc1[j]);
    }
  }
}

extern "C" void kernel_launch(void* const* d_in, const int* in_sizes, int n_in,
                              void* d_out, int out_size, void* d_ws, size_t ws_size,
                              hipStream_t stream) {
  const float* x    = (const float*)d_in[0];
  const float* w    = (const float*)d_in[1];
  const int*   perm = (const int*)d_in[2];
  float* out = (float*)d_out;
  int*   inv = (int*)d_ws;                 // 1 MiB scratch for inverse permutation

  const int n = KHEADS * NPER;
  invperm_kernel<<<(n + 255) / 256, 256, 0, stream>>>(perm, inv, n);

  dim3 grid(NPER / 128, KHEADS);           // 128 row-blocks x 16 heads
  size_t lds_bytes = (size_t)S2DIM * LDSW2 * sizeof(v2f);  // 273 KiB
  gemm_tanh_perm_kernel<<<grid, 256, lds_bytes, stream>>>(x, w, inv, out);
}